// SPAPooling_49426483642536
// MI455X (gfx1250) — compile-verified
//
#include <hip/hip_runtime.h>
#include <hip/hip_bf16.h>

typedef __attribute__((ext_vector_type(2))) float v2f;
typedef __attribute__((ext_vector_type(8))) float v8f;

#define NN 50000
#define FF 256
#define NO 64
#define DA 64
#define KK 256
#define EE 800000

__device__ inline v8f wmma_f32(v2f a, v2f b, v8f c) {
    // D = A(16x4,f32) * B(4x16,f32) + C(16x16,f32)
    return __builtin_amdgcn_wmma_f32_16x16x4_f32(
        false, a, false, b, (short)0, c, false, false);
}

// CDNA5 async global->LDS copy (ASYNCcnt-tracked), 16 bytes per lane.
// the LDS destination address is taken from a VGPR; low 32 bits of a generic shared pointer
// are exactly the LDS byte offset (aperture truncation rule).
__device__ inline void async_g2l_b128(void* lds_ptr, const void* gptr) {
    unsigned loff = (unsigned)(size_t)lds_ptr;
    unsigned long long ga = (unsigned long long)(size_t)gptr;
    asm volatile("global_load_async_to_lds_b128 %0, %1, off"
                 :: "v"(loff), "v"(ga) : "memory");
}
__device__ inline void wait_async0() {
    asm volatile("s_wait_asynccnt 0x0" ::: "memory");
}

// ---------------- generic row-major GEMM, one wave = one 16x16 C tile -------
// C[row0+.., col0+..] = A(MxKdim) @ B(KdimxNcols) + bias(col)
__global__ void k_gemm_bias(const float* __restrict__ A, const float* __restrict__ B,
                            const float* __restrict__ bias, float* __restrict__ C,
                            int lda, int ldb, int ldc, int Kdim) {
    const int row0 = blockIdx.x * 16;
    const int col0 = blockIdx.y * 16;
    const int lane = threadIdx.x & 31;
    const int m    = lane & 15;
    const int kh   = (lane >> 4) << 1;      // 0 or 2
    v8f acc = {};
    for (int k0 = 0; k0 < Kdim; k0 += 4) {
        v2f a, b;
        a.x = A[(row0 + m) * lda + k0 + kh];
        a.y = A[(row0 + m) * lda + k0 + kh + 1];
        b.x = B[(k0 + kh) * ldb + col0 + m];
        b.y = B[(k0 + kh + 1) * ldb + col0 + m];
        acc = wmma_f32(a, b, acc);
    }
    const int roff = (lane >> 4) << 3;
    const float bb = bias ? bias[col0 + m] : 0.0f;
#pragma unroll
    for (int r = 0; r < 8; ++r)
        C[(row0 + r + roff) * ldc + col0 + m] = acc[r] + bb;
}

// logits = scale * q @ kk^T   (q: N x DA row-major, kk: K x DA row-major)
__global__ void k_logits(const float* __restrict__ q, const float* __restrict__ kk,
                         float* __restrict__ S) {
    const int row0 = blockIdx.x * 16;
    const int col0 = blockIdx.y * 16;
    const int lane = threadIdx.x & 31;
    const int m    = lane & 15;
    const int kh   = (lane >> 4) << 1;
    v8f acc = {};
    for (int k0 = 0; k0 < DA; k0 += 4) {
        v2f a, b;
        a.x = q[(row0 + m) * DA + k0 + kh];
        a.y = q[(row0 + m) * DA + k0 + kh + 1];
        b.x = kk[(col0 + m) * DA + k0 + kh];        // B[d,c] = kk[c,d]
        b.y = kk[(col0 + m) * DA + k0 + kh + 1];
        acc = wmma_f32(a, b, acc);
    }
    const int roff = (lane >> 4) << 3;
#pragma unroll
    for (int r = 0; r < 8; ++r)
        S[(row0 + r + roff) * KK + col0 + m] = acc[r] * 0.125f; // 1/sqrt(64)
}

// ---------------------------------------------------------------------------
// LDS-staged 256x256 pooled GEMM:  C += sum_r Arow[amap(r)] (outer) Brow[bmap(r)]
// One workgroup (16 waves, 512 thr) owns the full 256x256 output; wave w owns
// the 16x256 strip (16 v8f accumulators).  32 reduction rows of A and B
// (64 KB) are staged per iteration via async global->LDS (ASYNCcnt), then
// each A fragment is reused across 16 WMMAs from LDS.  Rows >= Rtotal are
// zero-filled so WMMA control flow stays wave-uniform.
// ---------------------------------------------------------------------------
__global__ void k_pool_gemm(const float* __restrict__ Arows,
                            const float* __restrict__ Brows,
                            const int* __restrict__ amap,
                            const int* __restrict__ bmap,
                            float* __restrict__ C, int Rtotal, int chunk) {
    __shared__ float lsA[32 * 256];
    __shared__ float lsB[32 * 256];

    const int tid  = threadIdx.x;           // 0..511
    const int lane = tid & 31;
    const int wave = tid >> 5;              // 0..15
    const int m    = lane & 15;
    const int kh   = (lane >> 4) << 1;      // 0 or 2
    const int roff = (lane >> 4) << 3;      // 0 or 8
    const int row0 = wave * 16;
    const int rbase = blockIdx.x * chunk;

    v8f acc[16];
#pragma unroll
    for (int t = 0; t < 16; ++t) acc[t] = (v8f){};

    const int iters = chunk >> 5;           // chunk / 32
    for (int it = 0; it < iters; ++it) {
        const int r0 = rbase + (it << 5);
        // ---- stage 32 A-rows + 32 B-rows (2 * 32 * 1KB) into LDS ----------
        for (int c = tid; c < 4096; c += 512) {     // 16B chunks
            const int mat = c >> 11;                 // 0=A, 1=B
            const int r   = (c >> 6) & 31;           // staged row
            const int seg = c & 63;                  // 16B segment in row
            const int rg  = r0 + r;                  // global reduction index
            float* lrow = (mat ? lsB : lsA) + r * 256 + seg * 4;
            if (rg < Rtotal) {
                const int* map = mat ? bmap : amap;
                const int row  = map ? map[rg] : rg;
                const float* g = (mat ? Brows : Arows) + (size_t)row * 256 + seg * 4;
                async_g2l_b128(lrow, g);
            } else {
                lrow[0] = 0.f; lrow[1] = 0.f; lrow[2] = 0.f; lrow[3] = 0.f;
            }
        }
        wait_async0();
        __syncthreads();

        // ---- 8 k-steps of 4 rows; A fragment reused across 16 col tiles ---
        for (int ks = 0; ks < 8; ++ks) {
            const int k4 = ks << 2;
            v2f a;
            a.x = lsA[(k4 + kh) * 256 + row0 + m];
            a.y = lsA[(k4 + kh + 1) * 256 + row0 + m];
#pragma unroll
            for (int t = 0; t < 16; ++t) {
                v2f b;
                b.x = lsB[(k4 + kh) * 256 + t * 16 + m];
                b.y = lsB[(k4 + kh + 1) * 256 + t * 16 + m];
                acc[t] = wmma_f32(a, b, acc[t]);
            }
        }
        __syncthreads();
    }

#pragma unroll
    for (int t = 0; t < 16; ++t)
#pragma unroll
        for (int r = 0; r < 8; ++r)
            atomicAdd(&C[(row0 + r + roff) * KK + t * 16 + m], acc[t][r]);
}

// ----------------------------- scalar helpers -------------------------------
__global__ void k_zero(float* p, int n) {
    int i = blockIdx.x * blockDim.x + threadIdx.x;
    if (i < n) p[i] = 0.0f;
}

__global__ void k_deg_init(float* deg) {
    int i = blockIdx.x * blockDim.x + threadIdx.x;
    if (i < NN) deg[i] = 1.0f;
}

__global__ void k_deg_edges(const int* __restrict__ dst, float* deg) {
    int e = blockIdx.x * blockDim.x + threadIdx.x;
    if (e < EE) atomicAdd(&deg[dst[e]], 1.0f);
}

__global__ void k_dinv(const float* deg, float* dinv) {
    int i = blockIdx.x * blockDim.x + threadIdx.x;
    if (i < NN) dinv[i] = rsqrtf(deg[i]);
}

__global__ void k_edge_agg(const float* __restrict__ xw, const int* __restrict__ src,
                           const int* __restrict__ dst, const float* __restrict__ dinv,
                           float* __restrict__ agg) {
    int i = blockIdx.x * blockDim.x + threadIdx.x;    // over EE*NO
    if (i >= EE * NO) return;
    int e = i >> 6, f = i & 63;
    int s = src[e], d = dst[e];
    atomicAdd(&agg[d * NO + f], dinv[s] * dinv[d] * xw[s * NO + f]);
}

__global__ void k_emb(const float* __restrict__ agg, const float* __restrict__ xw,
                      const float* __restrict__ deg, float* __restrict__ xe) {
    int i = blockIdx.x * blockDim.x + threadIdx.x;    // over NN*NO
    if (i >= NN * NO) return;
    float v = agg[i] + xw[i] / deg[i >> 6];
    xe[i] = 1.0f / (1.0f + __expf(-v));
}

__global__ void k_pnorm(const float* p, float* pn) {
    __shared__ float red[NO];
    int t = threadIdx.x;
    red[t] = p[t] * p[t];
    __syncthreads();
    for (int s = NO / 2; s > 0; s >>= 1) {
        if (t < s) red[t] += red[t + s];
        __syncthreads();
    }
    if (t == 0) pn[0] = sqrtf(red[0]);
}

__global__ void k_score(const float* __restrict__ xe, const float* __restrict__ p,
                        const float* pn, float* __restrict__ work) {
    int n = blockIdx.x * blockDim.x + threadIdx.x;
    if (n >= NN) return;
    float d = 0.0f;
    for (int f = 0; f < NO; ++f) d += xe[n * NO + f] * p[f];
    work[n] = tanhf(d / pn[0]);
}

// deterministic 256-pass argmax top-k (ties -> lowest index, matching lax.top_k)
__global__ void k_topk(float* work, int* perm, float* score_k_out) {
    __shared__ float bv[1024];
    __shared__ int   bi[1024];
    int t = threadIdx.x;
    for (int k = 0; k < KK; ++k) {
        float best = -3.0e38f;
        int   besti = 0x7fffffff;
        for (int i = t; i < NN; i += 1024) {
            float v = work[i];
            if (v > best || (v == best && i < besti)) { best = v; besti = i; }
        }
        bv[t] = best; bi[t] = besti;
        __syncthreads();
        for (int s = 512; s > 0; s >>= 1) {
            if (t < s) {
                if (bv[t + s] > bv[t] || (bv[t + s] == bv[t] && bi[t + s] < bi[t])) {
                    bv[t] = bv[t + s]; bi[t] = bi[t + s];
                }
            }
            __syncthreads();
        }
        if (t == 0) {
            perm[k] = bi[0];
            score_k_out[k] = bv[0];
            work[bi[0]] = -3.0e38f;
        }
        __syncthreads();
    }
}

__global__ void k_gather_xrep(const float* __restrict__ xe, const int* __restrict__ perm,
                              float* __restrict__ xrep) {
    int i = blockIdx.x * blockDim.x + threadIdx.x;    // KK*NO
    if (i >= KK * NO) return;
    int k = i >> 6, f = i & 63;
    xrep[i] = xe[perm[k] * NO + f];
}

// row-wise softmax (in place) + entropy-loss accumulation
__global__ void k_softmax_ent(float* __restrict__ S, float* ent_acc) {
    __shared__ float red[KK];
    const int n = blockIdx.x, t = threadIdx.x;
    float v = S[n * KK + t];
    red[t] = v; __syncthreads();
    for (int s = KK / 2; s > 0; s >>= 1) {
        if (t < s) red[t] = fmaxf(red[t], red[t + s]);
        __syncthreads();
    }
    float mx = red[0]; __syncthreads();
    float ev = __expf(v - mx);
    red[t] = ev; __syncthreads();
    for (int s = KK / 2; s > 0; s >>= 1) {
        if (t < s) red[t] += red[t + s];
        __syncthreads();
    }
    float sm = red[0]; __syncthreads();
    float p = ev / sm;
    S[n * KK + t] = p;
    red[t] = -p * logf(p + 1e-15f);
    __syncthreads();
    for (int s = KK / 2; s > 0; s >>= 1) {
        if (t < s) red[t] += red[t + s];
        __syncthreads();
    }
    if (t == 0) atomicAdd(ent_acc, red[0]);
}

// assemble losses + tail outputs.  cross = trace(A_pool)
__global__ void k_final(const float* __restrict__ StS, const float* __restrict__ A_pool,
                        const int* __restrict__ perm, const float* ent_acc,
                        float* out_batch, float* out_perm, float* out_losses) {
    __shared__ float red[KK];
    int t = threadIdx.x;
    float sq = 0.0f;
    for (int i = t; i < KK * KK; i += KK) { float v = StS[i]; sq += v * v; }
    red[t] = sq; __syncthreads();
    for (int s = KK / 2; s > 0; s >>= 1) {
        if (t < s) red[t] += red[t + s];
        __syncthreads();
    }
    float sqsum = red[0]; __syncthreads();
    red[t] = A_pool[t * KK + t]; __syncthreads();
    for (int s = KK / 2; s > 0; s >>= 1) {
        if (t < s) red[t] += red[t + s];
        __syncthreads();
    }
    float cross = red[0];
    out_batch[t] = 0.0f;
    out_perm[t]  = (float)perm[t];
    if (t == 0) {
        float fro2 = (float)EE - 2.0f * cross + sqsum;
        out_losses[0] = sqrtf(fmaxf(fro2, 0.0f)) / ((float)NN * (float)NN);
        out_losses[1] = ent_acc[0] / (float)NN;
    }
}

extern "C" void kernel_launch(void* const* d_in, const int* in_sizes, int n_in,
                              void* d_out, int out_size, void* d_ws, size_t ws_size,
                              hipStream_t stream) {
    const float* x     = (const float*)d_in[0];   // N x F
    const int*   eidx  = (const int*)d_in[1];     // 2 x E
    const float* W_gcn = (const float*)d_in[3];   // F x NO
    const float* b_gcn = (const float*)d_in[4];   // NO
    const float* p_sel = (const float*)d_in[5];   // NO
    const float* Wq    = (const float*)d_in[6];   // NO x DA
    const float* Wk    = (const float*)d_in[7];   // NO x DA
    const int* src = eidx;
    const int* dst = eidx + EE;

    float* out = (float*)d_out;
    float* o_out    = out;                 // K x F
    float* o_apool  = out + 65536;         // K x K
    float* o_batch  = out + 131072;        // K
    float* o_perm   = out + 131328;        // K
    float* o_scorek = out + 131584;        // K
    float* o_loss   = out + 131840;        // 2

    // workspace carve (floats)
    float* ws   = (float*)d_ws;
    float* xw   = ws;  ws += NN * NO;
    float* agg  = ws;  ws += NN * NO;
    float* xe   = ws;  ws += NN * NO;
    float* q    = ws;  ws += NN * NO;
    float* S    = ws;  ws += (size_t)NN * KK;
    float* deg  = ws;  ws += NN;
    float* dinv = ws;  ws += NN;
    float* work = ws;  ws += NN;
    float* kkb  = ws;  ws += KK * NO;
    float* xrep = ws;  ws += KK * NO;
    float* StS  = ws;  ws += KK * KK;
    float* pn   = ws;  ws += 1;
    float* ent  = ws;  ws += 1;
    int*   perm = (int*)ws;

    // ---- zero accumulators -------------------------------------------------
    k_zero<<<(131842 + 255) / 256, 256, 0, stream>>>(out, 131842);
    k_zero<<<(NN * NO + 255) / 256, 256, 0, stream>>>(agg, NN * NO);
    k_zero<<<(KK * KK + 255) / 256, 256, 0, stream>>>(StS, KK * KK);
    k_zero<<<1, 32, 0, stream>>>(ent, 1);

    // ---- 1) GCN embedding --------------------------------------------------
    k_deg_init<<<(NN + 255) / 256, 256, 0, stream>>>(deg);
    k_deg_edges<<<(EE + 255) / 256, 256, 0, stream>>>(dst, deg);
    k_dinv<<<(NN + 255) / 256, 256, 0, stream>>>(deg, dinv);
    k_gemm_bias<<<dim3(NN / 16, NO / 16), 32, 0, stream>>>(x, W_gcn, b_gcn, xw,
                                                           FF, NO, NO, FF);
    k_edge_agg<<<(EE * NO + 255) / 256, 256, 0, stream>>>(xw, src, dst, dinv, agg);
    k_emb<<<(NN * NO + 255) / 256, 256, 0, stream>>>(agg, xw, deg, xe);

    // ---- 2) top-k selection ------------------------------------------------
    k_pnorm<<<1, NO, 0, stream>>>(p_sel, pn);
    k_score<<<(NN + 255) / 256, 256, 0, stream>>>(xe, p_sel, pn, work);
    k_topk<<<1, 1024, 0, stream>>>(work, perm, o_scorek);
    k_gather_xrep<<<(KK * NO + 255) / 256, 256, 0, stream>>>(xe, perm, xrep);

    // ---- 3) attention association -----------------------------------------
    k_gemm_bias<<<dim3(NN / 16, DA / 16), 32, 0, stream>>>(xe, Wq, nullptr, q,
                                                           NO, DA, DA, NO);
    k_gemm_bias<<<dim3(KK / 16, DA / 16), 32, 0, stream>>>(xrep, Wk, nullptr, kkb,
                                                           NO, DA, DA, NO);
    k_logits<<<dim3(NN / 16, KK / 16), 32, 0, stream>>>(q, kkb, S);
    k_softmax_ent<<<NN, KK, 0, stream>>>(S, ent);

    // ---- 4) pooled GEMMs: LDS-staged (async G->LDS), full 256x256 per WG ---
    // out = S^T @ x
    k_pool_gemm<<<(NN + 511) / 512, 512, 0, stream>>>(S, x, nullptr, nullptr,
                                                      o_out, NN, 512);
    // StS = S^T @ S
    k_pool_gemm<<<(NN + 511) / 512, 512, 0, stream>>>(S, S, nullptr, nullptr,
                                                      StS, NN, 512);
    // A_pool = Ssrc^T @ Sdst   (gather-fed via edge index maps)
    k_pool_gemm<<<(EE + 2047) / 2048, 512, 0, stream>>>(S, S, src, dst,
                                                        o_apool, EE, 2048);

    // ---- 5) losses + tail outputs -----------------------------------------
    k_final<<<1, KK, 0, stream>>>(StS, o_apool, perm, ent, o_batch, o_perm, o_loss);
}